// MultiHeadAttentionLayer_17506286698742
// MI455X (gfx1250) — compile-verified
//
#include <hip/hip_runtime.h>

typedef __attribute__((ext_vector_type(2))) float v2f;
typedef __attribute__((ext_vector_type(4))) float v4f;
typedef __attribute__((ext_vector_type(8))) float v8f;

#define GN 50000
#define GE 625000
#define GH 8
#define GD 16
// IN = H*D = 128 columns everywhere

// ---------------------------------------------------------------------------
// Persistent-block WMMA f32 GEMM:  Y[M,128] = X[M,128] @ W[128,128] + bias
// Block = 256 threads = 8 waves; each wave owns a 16-column panel of W in
// registers (32 k-steps x v2f) and computes 16x16 output tiles with
// V_WMMA_F32_16X16X4_F32. X tile (16x128) staged in LDS per M-tile.
// ---------------------------------------------------------------------------
__global__ __launch_bounds__(256) void proj_gemm_kernel(
    const float* __restrict__ X, const float* __restrict__ W,
    const float* __restrict__ bias, float* __restrict__ Y, int M)
{
  __shared__ float lds_x[16 * 128];

  const int t    = threadIdx.x;
  const int wave = t >> 5;        // 0..7 -> column tile
  const int lane = t & 31;
  const int n0   = wave << 4;     // column base
  const int half = lane >> 4;     // 0 | 1
  const int l15  = lane & 15;

  // --- Preload this wave's B panel of W: 32 k-steps of a 4x16 fragment.
  // B 4x16 layout (mirrors 32-bit A): lanes 0-15 hold K={k0,k0+1},
  // lanes 16-31 hold K={k0+2,k0+3}; N = lane&15 within the panel.
  v2f bfrag[32];
#pragma unroll
  for (int kk = 0; kk < 32; ++kk) {
    const int k0 = (kk << 2) + (half << 1);
    bfrag[kk].x = W[(k0 + 0) * 128 + n0 + l15];
    bfrag[kk].y = W[(k0 + 1) * 128 + n0 + l15];
  }
  const float bia = bias[n0 + l15];

  const int ntiles = (M + 15) >> 4;
  for (int tile = blockIdx.x; tile < ntiles; tile += gridDim.x) {
    const int m0 = tile << 4;

    __syncthreads();  // protect LDS from previous iteration's readers
    // Cooperative load of 16x128 X tile: 512 float4, 2 per thread.
#pragma unroll
    for (int i = 0; i < 2; ++i) {
      const int idx  = t + (i << 8);   // 0..511
      const int row  = idx >> 5;       // 32 float4 per row
      const int col4 = idx & 31;
      int gr = m0 + row;
      if (gr >= M) gr = M - 1;         // clamp rows of partial last tile
      const v4f v = *(const v4f*)(X + (size_t)gr * 128 + (col4 << 2));
      *(v4f*)(&lds_x[row * 128 + (col4 << 2)]) = v;
    }
    __syncthreads();

    v8f acc = {};
#pragma unroll
    for (int kk = 0; kk < 32; ++kk) {
      const int k0 = (kk << 2) + (half << 1);
      // A 16x4 f32 layout: lanes 0-15 M=lane, VGPR0=K0/VGPR1=K1;
      // lanes 16-31 M=lane-16, VGPR0=K2/VGPR1=K3.
      v2f a;
      a.x = lds_x[l15 * 128 + k0 + 0];
      a.y = lds_x[l15 * 128 + k0 + 1];
      acc = __builtin_amdgcn_wmma_f32_16x16x4_f32(
          /*neg_a=*/false, a, /*neg_b=*/false, bfrag[kk],
          /*c_mod=*/(short)0, acc, /*reuse_a=*/false, /*reuse_b=*/false);
    }

    // C/D layout: VGPR r -> row m0 + r + 8*half, col n0 + (lane&15)
#pragma unroll
    for (int r = 0; r < 8; ++r) {
      const int row = m0 + r + (half << 3);
      if (row < M)
        Y[(size_t)row * 128 + n0 + l15] = acc[r] + bia;
    }
  }
}

// ---------------------------------------------------------------------------
// Fused edge kernel: score = K[src]*Q[dst]*(1/4)*pe ; e_out <- score (in
// place over pe); s = exp(clamp(sum_d score,-5,5)) via 16-lane butterfly;
// atomic scatter of V[src]*s into hacc (h_out region) and s into z.
// 128 threads per edge, 2 edges per 256-thread block.
// ---------------------------------------------------------------------------
__global__ __launch_bounds__(256) void edge_attn_kernel(
    const float* __restrict__ Q, const float* __restrict__ K,
    const float* __restrict__ V, const int* __restrict__ src,
    const int* __restrict__ dst, float* __restrict__ e_out,
    float* __restrict__ hacc, float* __restrict__ z, int E)
{
  const size_t gid  = (size_t)blockIdx.x * 256 + threadIdx.x;
  const size_t edge = gid >> 7;
  if (edge >= (size_t)E) return;
  const int within = (int)(gid & 127);   // 0..127 = head*16 + d
  const int head   = within >> 4;

  const int s = src[edge];
  const int d = dst[edge];

  const float pe  = e_out[edge * 128 + within];
  const float kq  = K[(size_t)s * 128 + within] * Q[(size_t)d * 128 + within];
  const float sc  = kq * 0.25f * pe;     // 1/sqrt(D), D=16
  e_out[edge * 128 + within] = sc;

  // sum over D=16 lanes (head groups are 16 contiguous lanes in a wave32)
  float sum = sc;
#pragma unroll
  for (int m = 1; m < 16; m <<= 1) sum += __shfl_xor(sum, m, 16);

  const float sv = expf(fminf(fmaxf(sum, -5.0f), 5.0f));

  atomicAdd(&hacc[(size_t)d * 128 + within],
            V[(size_t)s * 128 + within] * sv);
  if ((within & 15) == 0)
    atomicAdd(&z[(size_t)d * GH + head], sv);
}

// ---------------------------------------------------------------------------
// h_out = hacc / (z + 1e-6), in place on the h_out region of d_out.
// ---------------------------------------------------------------------------
__global__ __launch_bounds__(256) void finalize_kernel(
    float* __restrict__ hacc, const float* __restrict__ z, int n)
{
  const int i = blockIdx.x * 256 + threadIdx.x;
  if (i >= n) return;
  const int node = i >> 7;
  const int head = (i >> 4) & 7;
  hacc[i] = hacc[i] / (z[node * GH + head] + 1e-6f);
}

extern "C" void kernel_launch(void* const* d_in, const int* in_sizes, int n_in,
                              void* d_out, int out_size, void* d_ws, size_t ws_size,
                              hipStream_t stream) {
  const float* h   = (const float*)d_in[0];
  const float* e   = (const float*)d_in[1];
  const int*   src = (const int*)  d_in[2];
  const int*   dst = (const int*)  d_in[3];
  const float* Wq  = (const float*)d_in[4];
  const float* bq  = (const float*)d_in[5];
  const float* Wk  = (const float*)d_in[6];
  const float* bk  = (const float*)d_in[7];
  const float* Wv  = (const float*)d_in[8];
  const float* bv  = (const float*)d_in[9];
  const float* We  = (const float*)d_in[10];
  const float* be  = (const float*)d_in[11];

  float* out   = (float*)d_out;
  float* h_out = out;                        // [N,128] accumulator, then result
  float* e_out = out + (size_t)GN * 128;     // [E,128] pe, then score in place

  float* ws = (float*)d_ws;
  float* Qb = ws;                            // [N,128]
  float* Kb = Qb + (size_t)GN * 128;         // [N,128]
  float* Vb = Kb + (size_t)GN * 128;         // [N,128]
  float* zb = Vb + (size_t)GN * 128;         // [N,8]

  // Zero the scatter accumulators (graph-capturable memsets).
  hipMemsetAsync(h_out, 0, (size_t)GN * 128 * sizeof(float), stream);
  hipMemsetAsync(zb,    0, (size_t)GN * GH  * sizeof(float), stream);

  // Projections (WMMA f32 GEMMs). Persistent blocks cap W register-panel
  // reload traffic; 2048 blocks fill all WGPs many times over.
  const int gN = min((GN + 15) / 16, 2048);
  const int gE = min((GE + 15) / 16, 2048);
  proj_gemm_kernel<<<gN, 256, 0, stream>>>(h, Wq, bq, Qb, GN);
  proj_gemm_kernel<<<gN, 256, 0, stream>>>(h, Wk, bk, Kb, GN);
  proj_gemm_kernel<<<gN, 256, 0, stream>>>(h, Wv, bv, Vb, GN);
  proj_gemm_kernel<<<gE, 256, 0, stream>>>(e, We, be, e_out, GE);

  // Fused edge scoring + scatter: E*128 threads.
  const long long eThreads = (long long)GE * 128;
  const int eBlocks = (int)((eThreads + 255) / 256);
  edge_attn_kernel<<<eBlocks, 256, 0, stream>>>(Qb, Kb, Vb, src, dst,
                                                e_out, h_out, zb, GE);

  // Normalize h_out.
  const int nElems  = GN * 128;
  const int nBlocks = (nElems + 255) / 256;
  finalize_kernel<<<nBlocks, 256, 0, stream>>>(h_out, zb, nElems);
}